// GCN_Node_18081812316383
// MI455X (gfx1250) — compile-verified
//
#include <hip/hip_runtime.h>
#include <hip/hip_bf16.h>

typedef __attribute__((ext_vector_type(2))) float v2f;
typedef __attribute__((ext_vector_type(8))) float v8f;

// Hardware f32 atomic add (global_atomic_add_f32), agent scope, no CAS loop.
__device__ __forceinline__ void atomicAddF(float* p, float v) {
    __hip_atomic_fetch_add(p, v, __ATOMIC_RELAXED, __HIP_MEMORY_SCOPE_AGENT);
}

// ---------------- degree / normalization ----------------

__global__ void k_fill_deg(float* __restrict__ deg, int n) {
    int i = blockIdx.x * blockDim.x + threadIdx.x;
    if (i < n) deg[i] = 1.0f;  // self-loop contributes 1
}

__global__ void k_count_deg(const int* __restrict__ dst, float* __restrict__ deg, int e) {
    int i = blockIdx.x * blockDim.x + threadIdx.x;
    if (i < e) atomicAddF(&deg[dst[i]], 1.0f);
}

__global__ void k_rsqrt_inplace(float* __restrict__ deg, int n) {
    int i = blockIdx.x * blockDim.x + threadIdx.x;
    if (i < n) {
        float d = deg[i];
        deg[i] = (d > 0.0f) ? rsqrtf(d) : 0.0f;
    }
}

// ---------------- dense GEMM via V_WMMA_F32_16X16X4_F32 ----------------
// H[M x NCOL] = A[M x 128] * W[128 x NCOL]; M multiple of 16; NCOL in {64,128}.
// One block per 16-row tile; wave w owns the 16-wide column tile w.
// A-frag (16x4 f32): VGPR0: K+0 (lanes 0-15) / K+2 (lanes 16-31); VGPR1: K+1 / K+3.
// B-frag (4x16 f32): same K split across lane halves, N = lane%16.
// C/D (16x16 f32, 8 VGPRs): VGPR v -> row v (lanes 0-15), row v+8 (lanes 16-31).
template <int NCOL>
__global__ __launch_bounds__(32 * (NCOL / 16))
void k_gemm_wmma(const float* __restrict__ A, const float* __restrict__ W,
                 float* __restrict__ H) {
    constexpr int T = 32 * (NCOL / 16);
    __shared__ float xs[16 * 128];

    const int tid = threadIdx.x;
    const int m0  = blockIdx.x * 16;

    // Stage the 16x128 A tile (8 KB) into LDS, coalesced.
    for (int i = tid; i < 16 * 128; i += T)
        xs[i] = A[(size_t)(m0 + (i >> 7)) * 128 + (i & 127)];
    __syncthreads();

    const int wave = tid >> 5;
    const int lane = tid & 31;
    const int half = lane >> 4;   // 0: lanes 0-15, 1: lanes 16-31
    const int lr   = lane & 15;
    const int n0   = wave * 16;

    v8f c = {0.f, 0.f, 0.f, 0.f, 0.f, 0.f, 0.f, 0.f};

    for (int k = 0; k < 128; k += 4) {
        v2f a, b;
        a.x = xs[lr * 128 + k + 2 * half + 0];
        a.y = xs[lr * 128 + k + 2 * half + 1];
        b.x = W[(size_t)(k + 2 * half + 0) * NCOL + n0 + lr];
        b.y = W[(size_t)(k + 2 * half + 1) * NCOL + n0 + lr];
        c = __builtin_amdgcn_wmma_f32_16x16x4_f32(
                /*neg_a=*/false, a, /*neg_b=*/false, b,
                /*c_mod=*/(short)0, c, /*reuse_a=*/false, /*reuse_b=*/false);
    }

#pragma unroll
    for (int v = 0; v < 8; ++v) {
        int row = m0 + v + 8 * half;
        H[(size_t)row * NCOL + n0 + lr] = c[v];
    }
}

// ---------------- self-loop init: agg[i] = h[i] * dinv[node]^2 ----------------

template <int C>
__global__ void k_selfinit(const float* __restrict__ H, const float* __restrict__ dinv,
                           float* __restrict__ agg, int total) {
    int i = blockIdx.x * blockDim.x + threadIdx.x;
    if (i < total) {
        float di = dinv[i / C];
        agg[i] = H[i] * di * di;
    }
}

// ---------------- edge scatter: one wave per edge ----------------
// C=128: 32 lanes x float4 = one full 512 B feature row per wave.
__global__ void k_scatter128(const int* __restrict__ src, const int* __restrict__ dst,
                             const float* __restrict__ dinv, const float* __restrict__ H,
                             float* __restrict__ agg, int E) {
    int e    = blockIdx.x * (blockDim.x >> 5) + (threadIdx.x >> 5);
    int lane = threadIdx.x & 31;
    if (e < E) {
        int   s    = src[e];
        int   d    = dst[e];
        float norm = dinv[s] * dinv[d];
        float4 v = ((const float4*)(H + (size_t)s * 128))[lane];
        float* op = agg + (size_t)d * 128 + lane * 4;
        atomicAddF(op + 0, v.x * norm);
        atomicAddF(op + 1, v.y * norm);
        atomicAddF(op + 2, v.z * norm);
        atomicAddF(op + 3, v.w * norm);
    }
}

// C=64: 32 lanes x float2.
__global__ void k_scatter64(const int* __restrict__ src, const int* __restrict__ dst,
                            const float* __restrict__ dinv, const float* __restrict__ H,
                            float* __restrict__ agg, int E) {
    int e    = blockIdx.x * (blockDim.x >> 5) + (threadIdx.x >> 5);
    int lane = threadIdx.x & 31;
    if (e < E) {
        int   s    = src[e];
        int   d    = dst[e];
        float norm = dinv[s] * dinv[d];
        float2 v = ((const float2*)(H + (size_t)s * 64))[lane];
        float* op = agg + (size_t)d * 64 + lane * 2;
        atomicAddF(op + 0, v.x * norm);
        atomicAddF(op + 1, v.y * norm);
    }
}

// ---------------- finalize passes ----------------

__global__ void k_bias_relu128(float* __restrict__ agg, const float* __restrict__ b, int total) {
    int i = blockIdx.x * blockDim.x + threadIdx.x;
    if (i < total) agg[i] = fmaxf(agg[i] + b[i & 127], 0.0f);
}

__global__ void k_bias64(float* __restrict__ out, const float* __restrict__ b, int total) {
    int i = blockIdx.x * blockDim.x + threadIdx.x;
    if (i < total) out[i] = out[i] + b[i & 63];
}

// ---------------- host launcher ----------------

extern "C" void kernel_launch(void* const* d_in, const int* in_sizes, int n_in,
                              void* d_out, int out_size, void* d_ws, size_t ws_size,
                              hipStream_t stream) {
    const float* x  = (const float*)d_in[0];  // [N,128]
    const int*   ei = (const int*)d_in[1];    // [2,E] row-major
    const float* W1 = (const float*)d_in[2];  // [128,128]
    const float* b1 = (const float*)d_in[3];  // [128]
    const float* W2 = (const float*)d_in[4];  // [128,64]
    const float* b2 = (const float*)d_in[5];  // [64]

    const int N = in_sizes[0] / 128;  // 100000 (multiple of 16)
    const int E = in_sizes[1] / 2;    // 1600000 (multiple of 8)
    const int* src = ei;
    const int* dst = ei + E;

    // Workspace layout (~128.4 MB total): deg/dinv, h1, relu(agg1), h2
    char*  ws   = (char*)d_ws;
    float* deg  = (float*)ws;
    size_t off  = ((size_t)N * 4 + 255) & ~(size_t)255;
    float* h1   = (float*)(ws + off); off += (size_t)N * 128 * 4;
    float* ag1  = (float*)(ws + off); off += (size_t)N * 128 * 4;
    float* h2   = (float*)(ws + off);
    float* out  = (float*)d_out;

    const int TB = 256;

    // Degrees (with self-loops) -> dinv, in place
    k_fill_deg<<<(N + TB - 1) / TB, TB, 0, stream>>>(deg, N);
    k_count_deg<<<(E + TB - 1) / TB, TB, 0, stream>>>(dst, deg, E);
    k_rsqrt_inplace<<<(N + TB - 1) / TB, TB, 0, stream>>>(deg, N);

    // Layer 1: h1 = x @ W1 ; agg = D^-1/2 (A+I) D^-1/2 h1 ; relu(+b1)
    k_gemm_wmma<128><<<N / 16, 256, 0, stream>>>(x, W1, h1);
    k_selfinit<128><<<((N * 128) + TB - 1) / TB, TB, 0, stream>>>(h1, deg, ag1, N * 128);
    k_scatter128<<<E / 8, 256, 0, stream>>>(src, dst, deg, h1, ag1, E);
    k_bias_relu128<<<((N * 128) + TB - 1) / TB, TB, 0, stream>>>(ag1, b1, N * 128);

    // Layer 2: h2 = relu_out @ W2 ; agg into d_out ; + b2
    k_gemm_wmma<64><<<N / 16, 128, 0, stream>>>(ag1, W2, h2);
    k_selfinit<64><<<((N * 64) + TB - 1) / TB, TB, 0, stream>>>(h2, deg, out, N * 64);
    k_scatter64<<<E / 8, 256, 0, stream>>>(src, dst, deg, h2, out, E);
    k_bias64<<<((N * 64) + TB - 1) / TB, TB, 0, stream>>>(out, b2, N * 64);
}